// Rudy_13030930776415
// MI455X (gfx1250) — compile-verified
//
#include <hip/hip_runtime.h>

// ---------------- problem constants (match reference) ----------------
#define NUM_BINS     256
#define BIN_SZ       3.90625f          // (1000-0)/256
#define UNIT_H_CAP   1.5625f
#define UNIT_V_CAP   1.45f
#define RUDY_EPS     1e-12f

// ---------------- tiling parameters ----------------
#define KSLICES      50                // K-dimension split for GPU fill
#define CHUNK        256               // nets staged in LDS per iteration
#define WAVES_PER_WG 4                 // 128 threads, wave32
#define NTILES       4                 // 16x16 ty-tiles per wave (16x64 strip)

typedef float v2f __attribute__((ext_vector_type(2)));
typedef float v4f __attribute__((ext_vector_type(4)));
typedef float v8f __attribute__((ext_vector_type(8)));

// -------------------------------------------------------------------
// Kernel 0: zero the h/v accumulation maps in workspace
// -------------------------------------------------------------------
__global__ void rudy_zero_kernel(float* __restrict__ p, int n) {
    int i = blockIdx.x * blockDim.x + threadIdx.x;
    if (i < n) p[i] = 0.0f;
}

// -------------------------------------------------------------------
// Kernel 1: per-net bbox + RUDY weights -> netdata[8] per net
//   {xmin, xmax, ymin, ymax} {wh, wv, 0, 0}
// -------------------------------------------------------------------
__global__ void rudy_bbox_kernel(const float* __restrict__ pin_pos,
                                 const float* __restrict__ net_weights,
                                 const int*   __restrict__ netpin_start,
                                 const int*   __restrict__ flat_netpin,
                                 float*       __restrict__ netdata,
                                 int num_nets, int num_pins_total) {
    int n = blockIdx.x * blockDim.x + threadIdx.x;
    if (n >= num_nets) return;
    int s = netpin_start[n];
    int e = netpin_start[n + 1];
    float xmin = 3.4e38f, xmax = -3.4e38f, ymin = 3.4e38f, ymax = -3.4e38f;
    for (int p = s; p < e; ++p) {
        int pin = flat_netpin[p];
        float px = pin_pos[pin];
        float py = pin_pos[pin + num_pins_total];
        xmin = fminf(xmin, px); xmax = fmaxf(xmax, px);
        ymin = fminf(ymin, py); ymax = fmaxf(ymax, py);
    }
    float w  = net_weights[n];
    float dx = xmax - xmin;
    float dy = ymax - ymin;
    float wh = (dy > 0.0f) ? (w / fmaxf(dy, RUDY_EPS)) : 0.0f;
    float wv = (dx > 0.0f) ? (w / fmaxf(dx, RUDY_EPS)) : 0.0f;
    if (e <= s) { xmin = xmax = ymin = ymax = 0.0f; wh = wv = 0.0f; }
    v4f* out = (v4f*)(netdata + 8 * n);
    out[0] = (v4f){xmin, xmax, ymin, ymax};
    out[1] = (v4f){wh, wv, 0.0f, 0.0f};
}

// -------------------------------------------------------------------
// Kernel 2: the two GEMMs  h_map += (ox*wh)^T oy,  v_map += (ox*wv)^T oy
// M = N = 256 bins, K = num_nets, via V_WMMA_F32_16X16X4_F32.
// Each wave: a 16(M) x 64(N) strip (4 tiles) for BOTH maps, one K-slice.
// A-fragments are computed once per K-step and reused across 4 B-fragments
// -> 8 WMMAs per ~44 VALU ops instead of 2 per ~20.
// Workgroup: 4 waves cover the full 256-wide ty range of one tx strip.
// -------------------------------------------------------------------
__global__ __launch_bounds__(32 * WAVES_PER_WG)
void rudy_wmma_kernel(const float* __restrict__ netdata,
                      float* __restrict__ h_map,
                      float* __restrict__ v_map,
                      int num_nets) {
    __shared__ float lds[CHUNK * 8];           // 8 KB: CHUNK nets x 8 floats

    const int wg   = blockIdx.x;               // 16 * KSLICES blocks
    const int tx   = wg & 15;                  // x-tile strip: 0..15
    const int ks   = wg >> 4;                  // K-slice: 0..KSLICES-1
    const int lane = threadIdx.x & 31;
    const int wave = threadIdx.x >> 5;
    const int ty0  = wave * NTILES;            // first ty tile of this wave

    // K range for this slice, rounded up to whole chunks (tail zero-padded)
    int kPerSlice = (num_nets + KSLICES - 1) / KSLICES;
    kPerSlice = (kPerSlice + CHUNK - 1) / CHUNK * CHUNK;
    const int k0 = ks * kPerSlice;
    const int k1 = k0 + kPerSlice;

    // A-fragment row (bin-x) and B-fragment base column (bin-y) for this lane
    const float bxl  = (float)(tx * 16 + (lane & 15)) * BIN_SZ;
    const float bxr  = bxl + BIN_SZ;
    const float byl0 = (float)(ty0 * 16 + (lane & 15)) * BIN_SZ;
    const int   kb   = (lane >> 4) * 2;        // this lane's k-offset in 4-group

    v8f accH[NTILES];
    v8f accV[NTILES];
#pragma unroll
    for (int j = 0; j < NTILES; ++j) { accH[j] = (v8f){}; accV[j] = (v8f){}; }

    for (int cbase = k0; cbase < k1; cbase += CHUNK) {
        __syncthreads();                        // protect LDS from prev iter reads
        // cooperative stage: CHUNK nets * 2 float4 each = 512 float4 / 128 thr
        for (int i = threadIdx.x; i < CHUNK * 2; i += blockDim.x) {
            int net = cbase + (i >> 1);
            v4f val = (v4f){0.0f, 0.0f, 0.0f, 0.0f};
            if (net < num_nets)
                val = ((const v4f*)netdata)[net * 2 + (i & 1)];
            ((v4f*)lds)[i] = val;
        }
        __syncthreads();

        for (int c = 0; c < CHUNK; c += 4) {
            const float* d0 = lds + (c + kb) * 8;       // net kbase + kb
            const float* d1 = lds + (c + kb + 1) * 8;   // net kbase + kb + 1
            // A fragments: overlap with this lane's bin column, once per step
            float ox0 = fmaxf(fminf(d0[1], bxr) - fmaxf(d0[0], bxl), 0.0f);
            float ox1 = fmaxf(fminf(d1[1], bxr) - fmaxf(d1[0], bxl), 0.0f);
            v2f aH = {ox0 * d0[4], ox1 * d1[4]};        // ox * wh
            v2f aV = {ox0 * d0[5], ox1 * d1[5]};        // ox * wv
            float ymin0 = d0[2], ymax0 = d0[3];
            float ymin1 = d1[2], ymax1 = d1[3];
#pragma unroll
            for (int j = 0; j < NTILES; ++j) {
                float byl = byl0 + (float)(16 * j) * BIN_SZ;
                float byr = byl + BIN_SZ;
                float oy0 = fmaxf(fminf(ymax0, byr) - fmaxf(ymin0, byl), 0.0f);
                float oy1 = fmaxf(fminf(ymax1, byr) - fmaxf(ymin1, byl), 0.0f);
                v2f bF = {oy0, oy1};                    // oy (B shared by h & v)
                accH[j] = __builtin_amdgcn_wmma_f32_16x16x4_f32(
                    false, aH, false, bF, (short)0, accH[j], false, false);
                accV[j] = __builtin_amdgcn_wmma_f32_16x16x4_f32(
                    false, aV, false, bF, (short)0, accV[j], false, false);
            }
        }
    }

    // C/D layout: VGPR r, lane l -> M = r + 8*(l>>4), N = l&15
    const int X0 = tx * 16 + 8 * (lane >> 4);
#pragma unroll
    for (int j = 0; j < NTILES; ++j) {
        const int Y = (ty0 + j) * 16 + (lane & 15);
#pragma unroll
        for (int r = 0; r < 8; ++r) {
            atomicAdd(&h_map[(X0 + r) * NUM_BINS + Y], accH[j][r]);
            atomicAdd(&v_map[(X0 + r) * NUM_BINS + Y], accV[j][r]);
        }
    }
}

// -------------------------------------------------------------------
// Kernel 3: out[x,y] = max(|h*sh|, |v*sv|)
// -------------------------------------------------------------------
__global__ void rudy_combine_kernel(const float* __restrict__ h_map,
                                    const float* __restrict__ v_map,
                                    float* __restrict__ out, int n) {
    int i = blockIdx.x * blockDim.x + threadIdx.x;
    if (i >= n) return;
    const float sh = 1.0f / (BIN_SZ * BIN_SZ * UNIT_H_CAP);
    const float sv = 1.0f / (BIN_SZ * BIN_SZ * UNIT_V_CAP);
    out[i] = fmaxf(fabsf(h_map[i] * sh), fabsf(v_map[i] * sv));
}

// -------------------------------------------------------------------
extern "C" void kernel_launch(void* const* d_in, const int* in_sizes, int n_in,
                              void* d_out, int out_size, void* d_ws, size_t ws_size,
                              hipStream_t stream) {
    const float* pin_pos      = (const float*)d_in[0];
    const float* net_weights  = (const float*)d_in[1];
    const int*   netpin_start = (const int*)d_in[2];
    const int*   flat_netpin  = (const int*)d_in[3];

    const int num_nets        = in_sizes[1];        // net_weights count
    const int num_pins_total  = in_sizes[0] / 2;    // pin_pos holds x then y

    // workspace layout: h_map | v_map | netdata(8 floats/net)
    float* h_map   = (float*)d_ws;
    float* v_map   = h_map + NUM_BINS * NUM_BINS;
    float* netdata = v_map + NUM_BINS * NUM_BINS;

    const int map_elems = 2 * NUM_BINS * NUM_BINS;
    rudy_zero_kernel<<<(map_elems + 255) / 256, 256, 0, stream>>>(h_map, map_elems);

    rudy_bbox_kernel<<<(num_nets + 255) / 256, 256, 0, stream>>>(
        pin_pos, net_weights, netpin_start, flat_netpin,
        netdata, num_nets, num_pins_total);

    rudy_wmma_kernel<<<16 * KSLICES, 32 * WAVES_PER_WG, 0, stream>>>(
        netdata, h_map, v_map, num_nets);

    rudy_combine_kernel<<<(out_size + 255) / 256, 256, 0, stream>>>(
        h_map, v_map, (float*)d_out, out_size);
}